// GroupedQueryAttention_84756884619995
// MI455X (gfx1250) — compile-verified
//
#include <hip/hip_runtime.h>
#include <hip/hip_bf16.h>
#include <math.h>

#define B_   4
#define T_   2048
#define D_   576
#define H_   9
#define KV_  3
#define G_   3
#define HD_  64
#define KVD_ 192   // KV_*HD_

typedef __attribute__((ext_vector_type(2))) float v2f;
typedef __attribute__((ext_vector_type(8))) float v8f;
typedef int b128i __attribute__((vector_size(16)));   // matches builtin's pointee type

__device__ __forceinline__ v8f wmma4(v2f a, v2f b, v8f c) {
  // D = A(16x4,f32) * B(4x16,f32) + C(16x16,f32)
  return __builtin_amdgcn_wmma_f32_16x16x4_f32(false, a, false, b, (short)0, c, false, false);
}

// 16-byte global -> LDS copy. Prefers the gfx1250 async-to-LDS data mover
// (ASYNCcnt path, no VGPR round trip); falls back to a plain vector copy.
__device__ __forceinline__ void copy16_g2l(float* lds, const float* g) {
#if __has_builtin(__builtin_amdgcn_global_load_async_to_lds_b128)
  __builtin_amdgcn_global_load_async_to_lds_b128(
      (__attribute__((address_space(1))) b128i*)(float*)g,
      (__attribute__((address_space(3))) b128i*)lds, 0, 0);
#else
  *(float4*)lds = *(const float4*)g;
#endif
}

__device__ __forceinline__ void wait_async_copies() {
#if __has_builtin(__builtin_amdgcn_global_load_async_to_lds_b128)
#if __has_builtin(__builtin_amdgcn_s_wait_asynccnt)
  __builtin_amdgcn_s_wait_asynccnt(0);
#else
  asm volatile("s_wait_asynccnt 0" ::: "memory");
#endif
#endif
}

// ---------------------------------------------------------------------------
// f32 WMMA GEMM: C[M,N] = A[M,K] @ B[K,N]   (M%64==0, N%16==0, K%4==0, K<=576)
// Block = 4 waves = 4 M-tiles sharing one 16-column B panel staged in LDS.
// ---------------------------------------------------------------------------
__global__ void __launch_bounds__(128) gemm_wmma_f32(const float* __restrict__ A,
                                                     const float* __restrict__ Bm,
                                                     float* __restrict__ C,
                                                     int M, int N, int K) {
  __shared__ float ldsB[576 * 16];                  // B panel, row-major [K][16]
  const int tid  = threadIdx.x;
  const int wave = tid >> 5;
  const int lane = tid & 31;
  const int half = lane >> 4;
  const int l16  = lane & 15;
  const int n0 = blockIdx.x * 16;
  const int m0 = (blockIdx.y * 4 + wave) * 16;

  // Cooperative async stage of the B panel (K*16 floats, 16B chunks).
  for (int c = tid; c < K * 4; c += 128) {
    const int k = c >> 2, j0 = (c & 3) * 4;
    copy16_g2l(ldsB + k * 16 + j0, Bm + (size_t)k * N + n0 + j0);
  }
  wait_async_copies();
  __syncthreads();

  const float* arow = A + (size_t)(m0 + l16) * K;   // A fragment: row = lane%16
  v8f acc = {};
  for (int k0 = 0; k0 < K; k0 += 4) {
    const int ka = k0 + 2 * half;                   // K split across lane halves
    v2f a, b;
    a.x = arow[ka];
    a.y = arow[ka + 1];
    b.x = ldsB[ka * 16 + l16];
    b.y = ldsB[(ka + 1) * 16 + l16];
    acc = wmma4(a, b, acc);
  }
#pragma unroll
  for (int r = 0; r < 8; ++r)                        // C row = r + 8*half
    C[(size_t)(m0 + r + 8 * half) * N + n0 + l16] = acc[r];
}

// ---------------------------------------------------------------------------
// RoPE (reference's interleaved rotate_half with concat[freqs,freqs] cos/sin)
// qlin (B,T,H*HD) -> Q (B,H,T,HD), folds 1/sqrt(HD) scale
// ---------------------------------------------------------------------------
__global__ void rope_q_kernel(const float* __restrict__ qlin, float* __restrict__ Qb) {
  const size_t i = (size_t)blockIdx.x * blockDim.x + threadIdx.x;
  const int d = (int)(i % HD_);
  const int h = (int)((i / HD_) % H_);
  const int t = (int)((i / ((size_t)HD_ * H_)) % T_);
  const int b = (int)(i / ((size_t)HD_ * H_ * T_));
  const size_t base = (size_t)(b * T_ + t) * D_ + (size_t)h * HD_;
  const int j = d & 31;
  const float inv = powf(10000.0f, -(float)j * (1.0f / 32.0f));
  const float ang = (float)t * inv;
  const float cs = cosf(ang), sn = sinf(ang);
  const float v0 = qlin[base + d];
  const float other = (d < 32) ? qlin[base + 2 * j + 1] : qlin[base + 2 * j];
  const float out = (d < 32) ? (v0 * cs - other * sn) : (v0 * cs + other * sn);
  Qb[((size_t)(b * H_ + h) * T_ + t) * HD_ + d] = out * 0.125f;   // 1/sqrt(64)
}

// klin/vlin (B,T,KV*HD) -> K,V (B,KV,T,HD); RoPE on K, plain transpose on V
__global__ void rope_kv_kernel(const float* __restrict__ klin, const float* __restrict__ vlin,
                               float* __restrict__ Kb, float* __restrict__ Vb) {
  const size_t i = (size_t)blockIdx.x * blockDim.x + threadIdx.x;
  const int d = (int)(i % HD_);
  const int n = (int)((i / HD_) % KV_);
  const int t = (int)((i / ((size_t)HD_ * KV_)) % T_);
  const int b = (int)(i / ((size_t)HD_ * KV_ * T_));
  const size_t base = (size_t)(b * T_ + t) * KVD_ + (size_t)n * HD_;
  const int j = d & 31;
  const float inv = powf(10000.0f, -(float)j * (1.0f / 32.0f));
  const float ang = (float)t * inv;
  const float cs = cosf(ang), sn = sinf(ang);
  const float v0 = klin[base + d];
  const float other = (d < 32) ? klin[base + 2 * j + 1] : klin[base + 2 * j];
  const float outk = (d < 32) ? (v0 * cs - other * sn) : (v0 * cs + other * sn);
  const size_t dst = ((size_t)(b * KV_ + n) * T_ + t) * HD_ + d;
  Kb[dst] = outk;
  Vb[dst] = vlin[base + d];
}

// ---------------------------------------------------------------------------
// Flash attention. Block = 4 waves = 4 consecutive 16-row Q tiles of one (b,h).
// K/V tiles staged once per block in LDS via async copies; causal online softmax.
// Q (B,H,T,HD) pre-scaled; K,V (B,KV,T,HD). ctx written as (B,T,H,HD)=(B,T,D).
// ---------------------------------------------------------------------------
__global__ void __launch_bounds__(128) attn_wmma(const float* __restrict__ Q,
                                                 const float* __restrict__ Kb,
                                                 const float* __restrict__ Vb,
                                                 float* __restrict__ ctx) {
  __shared__ float Kt[16 * HD_];      // staged K tile (rows = key idx, cols = hd)
  __shared__ float Vt[16 * HD_];      // staged V tile
  __shared__ float pbuf[4][16][17];   // per-wave P staging (C-layout -> A-layout)
  const int tid  = threadIdx.x;
  const int wave = tid >> 5;
  const int lane = tid & 31;
  const int half = lane >> 4;
  const int l16  = lane & 15;
  const int bh = blockIdx.x;
  const int b = bh / H_, h = bh % H_;
  const int n = h / G_;                        // kv head
  const int qt    = blockIdx.y * 4 + wave;     // this wave's q tile
  const int qtmax = blockIdx.y * 4 + 3;        // block's last q tile

  const float* Qh = Q  + (size_t)(b * H_  + h) * T_ * HD_;
  const float* Kh = Kb + (size_t)(b * KV_ + n) * T_ * HD_;
  const float* Vh = Vb + (size_t)(b * KV_ + n) * T_ * HD_;

  // Preload Q fragments for this wave's 16x64 tile (16 K-steps of 4)
  v2f qa[16];
  {
    const float* qrow = Qh + (size_t)(qt * 16 + l16) * HD_;
#pragma unroll
    for (int kk = 0; kk < 16; ++kk) {
      qa[kk].x = qrow[4 * kk + 2 * half];
      qa[kk].y = qrow[4 * kk + 2 * half + 1];
    }
  }

  v8f acc0 = {}, acc1 = {}, acc2 = {}, acc3 = {};   // ctx 16x64 as four 16x16
  float m_i[8], l_i[8];
#pragma unroll
  for (int r = 0; r < 8; ++r) { m_i[r] = -__builtin_inff(); l_i[r] = 0.0f; }

  for (int kt = 0; kt <= qtmax; ++kt) {
    // ---- cooperative stage of this K/V tile (2 x 4KB, 16B chunks, coalesced)
    const float* kg = Kh + (size_t)kt * 16 * HD_;
    const float* vg = Vh + (size_t)kt * 16 * HD_;
#pragma unroll
    for (int i = 0; i < 2; ++i) {
      const int c = (tid + i * 128) * 4;           // float offset of 16B chunk
      copy16_g2l(Kt + c, kg + c);
      copy16_g2l(Vt + c, vg + c);
    }
    if (kt < qtmax) {                              // prefetch next tiles
      __builtin_prefetch(kg + 16 * HD_, 0, 0);
      __builtin_prefetch(vg + 16 * HD_, 0, 0);
    }
    wait_async_copies();
    __syncthreads();

    if (kt <= qt) {                                // wave-uniform guard
      // ---- S = Q @ K^T (B-frag of K^T == row-major reads of staged K)
      const float* krow = Kt + l16 * HD_;
      v8f s = {};
#pragma unroll
      for (int kk = 0; kk < 16; ++kk) {
        v2f kb;
        kb.x = krow[4 * kk + 2 * half];
        kb.y = krow[4 * kk + 2 * half + 1];
        s = wmma4(qa[kk], kb, s);
      }
      // ---- causal mask (diagonal tile only): col l16, row r+8*half
      if (kt == qt) {
#pragma unroll
        for (int r = 0; r < 8; ++r)
          if (l16 > r + 8 * half) s[r] = -__builtin_inff();
      }
      // ---- online softmax: row reductions across the 16-lane half
      float corr[8];
#pragma unroll
      for (int r = 0; r < 8; ++r) {
        float mx = s[r];
#pragma unroll
        for (int off = 1; off < 16; off <<= 1) mx = fmaxf(mx, __shfl_xor(mx, off, 32));
        const float mnew = fmaxf(m_i[r], mx);
        const float p = expf(s[r] - mnew);
        float sum = p;
#pragma unroll
        for (int off = 1; off < 16; off <<= 1) sum += __shfl_xor(sum, off, 32);
        corr[r] = expf(m_i[r] - mnew);
        l_i[r]  = l_i[r] * corr[r] + sum;
        m_i[r]  = mnew;
        pbuf[wave][r + 8 * half][l16] = p;          // stage P in C-layout
      }
#pragma unroll
      for (int r = 0; r < 8; ++r) {
        acc0[r] *= corr[r]; acc1[r] *= corr[r]; acc2[r] *= corr[r]; acc3[r] *= corr[r];
      }
      asm volatile("" ::: "memory");               // LDS stores before reads (same-wave DS in-order)
      // ---- ctx += P @ V  (P as A-frags, V B-frags, both from LDS)
#pragma unroll
      for (int c = 0; c < 4; ++c) {
        const int kk0 = 4 * c + 2 * half;
        v2f pa;
        pa.x = pbuf[wave][l16][kk0];
        pa.y = pbuf[wave][l16][kk0 + 1];
        v2f vb0, vb1, vb2, vb3;
        vb0.x = Vt[kk0 * HD_ +  0 + l16];  vb0.y = Vt[(kk0 + 1) * HD_ +  0 + l16];
        vb1.x = Vt[kk0 * HD_ + 16 + l16];  vb1.y = Vt[(kk0 + 1) * HD_ + 16 + l16];
        vb2.x = Vt[kk0 * HD_ + 32 + l16];  vb2.y = Vt[(kk0 + 1) * HD_ + 32 + l16];
        vb3.x = Vt[kk0 * HD_ + 48 + l16];  vb3.y = Vt[(kk0 + 1) * HD_ + 48 + l16];
        acc0 = wmma4(pa, vb0, acc0);
        acc1 = wmma4(pa, vb1, acc1);
        acc2 = wmma4(pa, vb2, acc2);
        acc3 = wmma4(pa, vb3, acc3);
      }
    }
    __syncthreads();                               // all reads done before restage
  }

  // ---- epilogue: ctx[b][qi][h][hd] = acc / l
#pragma unroll
  for (int r = 0; r < 8; ++r) {
    const int qi = qt * 16 + r + 8 * half;
    float* dst = ctx + ((size_t)(b * T_ + qi) * H_ + h) * HD_;
    const float il = 1.0f / l_i[r];
    dst[ 0 + l16] = acc0[r] * il;
    dst[16 + l16] = acc1[r] * il;
    dst[32 + l16] = acc2[r] * il;
    dst[48 + l16] = acc3[r] * il;
  }
}

// ---------------------------------------------------------------------------
extern "C" void kernel_launch(void* const* d_in, const int* in_sizes, int n_in,
                              void* d_out, int out_size, void* d_ws, size_t ws_size,
                              hipStream_t stream) {
  (void)in_sizes; (void)n_in; (void)out_size; (void)ws_size;
  const float* x   = (const float*)d_in[0];
  const float* w_q = (const float*)d_in[1];
  const float* w_k = (const float*)d_in[2];
  const float* w_v = (const float*)d_in[3];
  const float* w_o = (const float*)d_in[4];
  float* outp = (float*)d_out;

  float* ws = (float*)d_ws;
  const size_t QL = (size_t)B_ * T_ * D_;     // 4,718,592 floats
  const size_t KL = (size_t)B_ * T_ * KVD_;   // 1,572,864 floats
  float* qlin = ws;
  float* klin = ws + QL;
  float* vlin = ws + QL + KL;
  float* Qb   = ws + QL + 2 * KL;
  float* Kbf  = ws + 2 * QL + 2 * KL;
  float* Vbf  = ws + 2 * QL + 3 * KL;
  float* ctx  = qlin;                          // qlin dead after rope_q; reuse as ctx

  const int M = B_ * T_;                       // 8192 rows
  dim3 blk(128);

  // Projections: q = x@w_q, k = x@w_k, v = x@w_v
  gemm_wmma_f32<<<dim3(D_   / 16, M / 64), blk, 0, stream>>>(x, w_q, qlin, M, D_,   D_);
  gemm_wmma_f32<<<dim3(KVD_ / 16, M / 64), blk, 0, stream>>>(x, w_k, klin, M, KVD_, D_);
  gemm_wmma_f32<<<dim3(KVD_ / 16, M / 64), blk, 0, stream>>>(x, w_v, vlin, M, KVD_, D_);

  // RoPE + transpose to head-major layouts
  rope_q_kernel <<<(B_ * T_ * H_  * HD_) / 256, 256, 0, stream>>>(qlin, Qb);
  rope_kv_kernel<<<(B_ * T_ * KV_ * HD_) / 256, 256, 0, stream>>>(klin, vlin, Kbf, Vbf);

  // Causal flash attention -> ctx (B,T,H,HD) = (B,T,D)
  attn_wmma<<<dim3(B_ * H_, (T_ / 16) / 4), blk, 0, stream>>>(Qb, Kbf, Vbf, ctx);

  // Output projection
  gemm_wmma_f32<<<dim3(D_ / 16, M / 64), blk, 0, stream>>>(ctx, w_o, outp, M, D_, D_);
}